// Net2_80676665688184
// MI455X (gfx1250) — compile-verified
//
#include <hip/hip_runtime.h>
#include <hip/hip_bf16.h>
#include <math.h>

#define NHEADS 3
#define CDIM   32
#define HC     96
#define NEG_SLOPE 0.2f

typedef __attribute__((ext_vector_type(2))) float v2f;
typedef __attribute__((ext_vector_type(8))) float v8f;

// ---- order-preserving float <-> uint transform (for atomic segment-max) ----
__device__ __forceinline__ unsigned floatOrder(float f) {
    unsigned u = __float_as_uint(f);
    return (u & 0x80000000u) ? ~u : (u | 0x80000000u);
}
__device__ __forceinline__ float orderInv(unsigned u) {
    return __uint_as_float((u & 0x80000000u) ? (u ^ 0x80000000u) : ~u);
}

// =====================================================================
// Dual GEMM: OL = X*Wl + bl, OR = X*Wr + br     (X:[N,K], W:[K,96])
// One wave computes one 16x16 output tile of both GEMMs using
// V_WMMA_F32_16X16X4_F32 (full fp32 matrix path, K-step 4).
// A 16x4 fp32 fragment: lanes 0-15 hold M=lane, K=k0..k0+1 (v0,v1);
//                       lanes 16-31 hold K=k0+2..k0+3.   (ISA 7.12.2)
// =====================================================================
__global__ void gemm_dual_wmma(const float* __restrict__ X, int K,
                               const float* __restrict__ Wl, const float* __restrict__ bl,
                               const float* __restrict__ Wr, const float* __restrict__ br,
                               float* __restrict__ OL, float* __restrict__ OR_,
                               int mtiles)
{
    int wave = (int)((blockIdx.x * blockDim.x + threadIdx.x) >> 5);
    int lane = (int)(threadIdx.x & 31);
    const int ntiles = HC / 16;                 // 6 column tiles
    int tm = wave / ntiles;
    int tn = wave % ntiles;
    if (tm >= mtiles) return;                   // wave-uniform: EXEC stays all-ones

    int row  = tm * 16 + (lane & 15);
    int col  = tn * 16 + (lane & 15);
    int koff = (lane >> 4) * 2;                 // 0 or 2

    v8f cl = {0.f,0.f,0.f,0.f,0.f,0.f,0.f,0.f};
    v8f cr = {0.f,0.f,0.f,0.f,0.f,0.f,0.f,0.f};

    const float* xrow = X + (size_t)row * K + koff;
    for (int k = 0; k < K; k += 4) {
        v2f a = *(const v2f*)(xrow + k);        // 8B aligned (K and koff even)
        v2f b0; b0.x = Wl[(k + koff) * HC + col];
                b0.y = Wl[(k + koff + 1) * HC + col];
        cl = __builtin_amdgcn_wmma_f32_16x16x4_f32(false, a, false, b0,
                                                   (short)0, cl, false, false);
        v2f b1; b1.x = Wr[(k + koff) * HC + col];
                b1.y = Wr[(k + koff + 1) * HC + col];
        cr = __builtin_amdgcn_wmma_f32_16x16x4_f32(false, a, false, b1,
                                                   (short)0, cr, false, false);
    }

    // D layout: VGPR v -> row tm*16 + v + 8*(lane>>4), col tn*16 + (lane&15)
    int rbase = tm * 16 + (lane >> 4) * 8;
    float bL = bl[col], bR = br[col];
#pragma unroll
    for (int v = 0; v < 8; ++v) {
        OL [(size_t)(rbase + v) * HC + col] = cl[v] + bL;
        OR_[(size_t)(rbase + v) * HC + col] = cr[v] + bR;
    }
}

// =====================================================================
// Edge pass A: logit[e,h] = sum_c leaky_relu(xl[src]+xr[dst]) * att ;
//              segment-max into mbuf (order-transformed uint atomics).
// One wave per edge, lane == channel.
// =====================================================================
__global__ void edge_logit_max(const float* __restrict__ XL, const float* __restrict__ XR,
                               const float* __restrict__ att,
                               const int* __restrict__ src, const int* __restrict__ dst,
                               int E, int Efull,
                               float* __restrict__ logit, unsigned* __restrict__ mbuf)
{
    int wid  = (int)((blockIdx.x * blockDim.x + threadIdx.x) >> 5);
    int lane = (int)(threadIdx.x & 31);
    if (wid >= Efull) return;
    int s, d;
    if (wid < E) { s = src[wid]; d = dst[wid]; } else { s = wid - E; d = s; }  // self loops
    const float* xls = XL + (size_t)s * HC;
    const float* xrd = XR + (size_t)d * HC;
#pragma unroll
    for (int h = 0; h < NHEADS; ++h) {
        float e = xls[h * CDIM + lane] + xrd[h * CDIM + lane];
        e = e > 0.0f ? e : NEG_SLOPE * e;
        float p = e * att[h * CDIM + lane];
#pragma unroll
        for (int off = 16; off > 0; off >>= 1) p += __shfl_down(p, off, 32);
        if (lane == 0) {
            logit[(size_t)wid * NHEADS + h] = p;
            atomicMax(&mbuf[d * NHEADS + h], floatOrder(p));
        }
    }
}

// Edge pass B: exbuf = exp(logit - m[dst]) in place; denom[dst] += ex
__global__ void edge_exp_sum(float* __restrict__ exbuf, const unsigned* __restrict__ mbuf,
                             const int* __restrict__ dst, int E, int Efull,
                             float* __restrict__ den)
{
    int i = blockIdx.x * blockDim.x + threadIdx.x;
    int tot = Efull * NHEADS;
    if (i >= tot) return;
    int eid = i / NHEADS;
    int h   = i - eid * NHEADS;
    int d   = (eid < E) ? dst[eid] : (eid - E);
    float mv = orderInv(mbuf[d * NHEADS + h]);
    float v  = expf(exbuf[i] - mv);
    exbuf[i] = v;
    atomicAdd(&den[d * NHEADS + h], v);
}

// Edge pass C: alpha = ex/denom[dst]; agg[dst] += xl[src]*alpha
// (optionally writes alpha back in place for the alpha2 output)
__global__ void edge_aggregate(const float* __restrict__ XL, float* __restrict__ exbuf,
                               const float* __restrict__ den,
                               const int* __restrict__ src, const int* __restrict__ dst,
                               int E, int Efull,
                               float* __restrict__ agg, int write_alpha)
{
    int wid  = (int)((blockIdx.x * blockDim.x + threadIdx.x) >> 5);
    int lane = (int)(threadIdx.x & 31);
    if (wid >= Efull) return;
    int s, d;
    if (wid < E) { s = src[wid]; d = dst[wid]; } else { s = wid - E; d = s; }
    const float* xls = XL + (size_t)s * HC;
    float* aggd = agg + (size_t)d * HC;
#pragma unroll
    for (int h = 0; h < NHEADS; ++h) {
        float ex    = exbuf[(size_t)wid * NHEADS + h];   // broadcast load
        float alpha = ex / den[d * NHEADS + h];
        if (write_alpha && lane == 0) exbuf[(size_t)wid * NHEADS + h] = alpha;
        atomicAdd(&aggd[h * CDIM + lane], xls[h * CDIM + lane] * alpha);
    }
}

// out = tanh(agg + bias)   (agg may alias out)
__global__ void bias_tanh(const float* __restrict__ agg, const float* __restrict__ bias,
                          float* __restrict__ out, int total)
{
    int i = blockIdx.x * blockDim.x + threadIdx.x;
    if (i >= total) return;
    out[i] = tanhf(agg[i] + bias[i % HC]);
}

// ---- fills ----
__global__ void fill_f32(float* p, float v, int n)    { int i = blockIdx.x*blockDim.x+threadIdx.x; if (i < n) p[i] = v; }
__global__ void fill_u32(unsigned* p, unsigned v, int n){ int i = blockIdx.x*blockDim.x+threadIdx.x; if (i < n) p[i] = v; }
__global__ void fill_i32(int* p, int v, int n)        { int i = blockIdx.x*blockDim.x+threadIdx.x; if (i < n) p[i] = v; }

// first node per graph (segment_min of node index over batch)
__global__ void seg_first(const int* __restrict__ batch, int* __restrict__ idx, int N)
{
    int i = blockIdx.x * blockDim.x + threadIdx.x;
    if (i < N) atomicMin(&idx[batch[i]], i);
}

// readout MLP: feat=[h1|h2] (192) -> relu(192x128) -> 128x2 -> log_softmax
__global__ void readout_mlp(const float* __restrict__ H1, const float* __restrict__ H2,
                            const int* __restrict__ idx,
                            const float* __restrict__ W1, const float* __restrict__ b1,
                            const float* __restrict__ W2, const float* __restrict__ b2,
                            float* __restrict__ out)
{
    __shared__ float feat[2 * HC];
    __shared__ float hid[128];
    int g = blockIdx.x, t = threadIdx.x;
    int node = idx[g];
    for (int k = t; k < 2 * HC; k += blockDim.x)
        feat[k] = (k < HC) ? H1[(size_t)node * HC + k] : H2[(size_t)node * HC + (k - HC)];
    __syncthreads();
    float acc = b1[t];
    for (int k = 0; k < 2 * HC; ++k) acc += feat[k] * W1[k * 128 + t];
    hid[t] = acc > 0.f ? acc : 0.f;
    __syncthreads();
    if (t == 0) {
        float l0 = b2[0], l1 = b2[1];
        for (int k = 0; k < 128; ++k) { l0 += hid[k] * W2[2 * k]; l1 += hid[k] * W2[2 * k + 1]; }
        float mx  = fmaxf(l0, l1);
        float lse = mx + logf(expf(l0 - mx) + expf(l1 - mx));
        out[g * 2 + 0] = l0 - lse;
        out[g * 2 + 1] = l1 - lse;
    }
}

static inline int cdiv(long long a, long long b) { return (int)((a + b - 1) / b); }

extern "C" void kernel_launch(void* const* d_in, const int* in_sizes, int n_in,
                              void* d_out, int out_size, void* d_ws, size_t ws_size,
                              hipStream_t stream)
{
    (void)n_in; (void)ws_size;
    const float* x     = (const float*)d_in[0];
    const int*   eidx  = (const int*)  d_in[1];
    const int*   batch = (const int*)  d_in[2];
    const float* Wl1   = (const float*)d_in[3];
    const float* bl1   = (const float*)d_in[4];
    const float* Wr1   = (const float*)d_in[5];
    const float* br1   = (const float*)d_in[6];
    const float* att1  = (const float*)d_in[7];
    const float* bias1 = (const float*)d_in[8];
    const float* Wl2   = (const float*)d_in[9];
    const float* bl2   = (const float*)d_in[10];
    const float* Wr2   = (const float*)d_in[11];
    const float* br2   = (const float*)d_in[12];
    const float* att2  = (const float*)d_in[13];
    const float* bias2 = (const float*)d_in[14];
    const float* W1    = (const float*)d_in[15];
    const float* b1    = (const float*)d_in[16];
    const float* W2    = (const float*)d_in[17];
    const float* b2    = (const float*)d_in[18];

    const int IN_DIM = 128;
    int N     = in_sizes[0] / IN_DIM;                       // 100000
    int E     = in_sizes[1] / 2;                            // 1600000
    int Efull = E + N;                                      // + self loops
    int G     = (out_size - N * HC - Efull * NHEADS) / 2;   // 100

    const int* src = eidx;
    const int* dst = eidx + E;

    // workspace layout
    float*    XL  = (float*)d_ws;                 // [N,96]
    float*    XR  = XL + (size_t)N * HC;          // [N,96] (also reused as layer-1 agg)
    float*    H1  = XR + (size_t)N * HC;          // [N,96]
    unsigned* M   = (unsigned*)(H1 + (size_t)N * HC);  // [N,3] ordered-uint max
    float*    DEN = (float*)(M + (size_t)N * NHEADS);  // [N,3]
    int*      IDX = (int*)(DEN + (size_t)N * NHEADS);  // [G]

    float* outH2    = (float*)d_out;                      // [N,96]
    float* outAlpha = outH2 + (size_t)N * HC;             // [Efull,3] (also logit/ex scratch)
    float* outLog   = outAlpha + (size_t)Efull * NHEADS;  // [G,2]

    const int BT = 256;
    int mtiles      = N / 16;                                  // 6250
    int blocks_gemm = cdiv((long long)mtiles * (HC / 16) * 32, BT);
    int blocks_edge = cdiv((long long)Efull * 32, BT);
    int blocks_eh   = cdiv((long long)Efull * NHEADS, BT);
    int blocks_nh   = cdiv((long long)N * NHEADS, BT);
    int blocks_nf   = cdiv((long long)N * HC, BT);

    // ================= Layer 1 =================
    gemm_dual_wmma<<<blocks_gemm, BT, 0, stream>>>(x, IN_DIM, Wl1, bl1, Wr1, br1, XL, XR, mtiles);
    fill_u32<<<blocks_nh, BT, 0, stream>>>(M, 0u, N * NHEADS);
    fill_f32<<<blocks_nh, BT, 0, stream>>>(DEN, 0.f, N * NHEADS);
    edge_logit_max<<<blocks_edge, BT, 0, stream>>>(XL, XR, att1, src, dst, E, Efull, outAlpha, M);
    edge_exp_sum<<<blocks_eh, BT, 0, stream>>>(outAlpha, M, dst, E, Efull, DEN);
    fill_f32<<<blocks_nf, BT, 0, stream>>>(XR, 0.f, N * HC);          // XR becomes aggregator
    edge_aggregate<<<blocks_edge, BT, 0, stream>>>(XL, outAlpha, DEN, src, dst, E, Efull, XR, 0);
    bias_tanh<<<blocks_nf, BT, 0, stream>>>(XR, bias1, H1, N * HC);

    // ================= Layer 2 =================
    gemm_dual_wmma<<<blocks_gemm, BT, 0, stream>>>(H1, HC, Wl2, bl2, Wr2, br2, XL, XR, mtiles);
    fill_u32<<<blocks_nh, BT, 0, stream>>>(M, 0u, N * NHEADS);
    fill_f32<<<blocks_nh, BT, 0, stream>>>(DEN, 0.f, N * NHEADS);
    edge_logit_max<<<blocks_edge, BT, 0, stream>>>(XL, XR, att2, src, dst, E, Efull, outAlpha, M);
    edge_exp_sum<<<blocks_eh, BT, 0, stream>>>(outAlpha, M, dst, E, Efull, DEN);
    fill_f32<<<blocks_nf, BT, 0, stream>>>(outH2, 0.f, N * HC);
    edge_aggregate<<<blocks_edge, BT, 0, stream>>>(XL, outAlpha, DEN, src, dst, E, Efull, outH2, 1);
    bias_tanh<<<blocks_nf, BT, 0, stream>>>(outH2, bias2, outH2, N * HC);

    // ================= Readout =================
    fill_i32<<<cdiv(G, BT), BT, 0, stream>>>(IDX, 0x7FFFFFFF, G);
    seg_first<<<cdiv(N, BT), BT, 0, stream>>>(batch, IDX, N);
    readout_mlp<<<G, 128, 0, stream>>>(H1, outH2, IDX, W1, b1, W2, b2, outLog);
}